// MultiGrapher_70351564309002
// MI455X (gfx1250) — compile-verified
//
#include <hip/hip_runtime.h>
#include <math.h>

// Problem constants (from the reference)
#define BATCH  4
#define CCH    192
#define HDIM   64
#define WDIM   64
#define KNN_K  9
#define NODES1 (HDIM * WDIM)   // 4096
#define NODES2 (NODES1 / 4)    // 1024
#define NODES3 (NODES2 / 4)    // 256
#define LDS_STRIDE 200         // 192 + pad: breaks 64-bank conflict on row accesses

typedef __attribute__((ext_vector_type(2))) float v2f;
typedef __attribute__((ext_vector_type(8))) float v8f;

// ---------------------------------------------------------------------------
// WMMA fragment helper: acc += A(16xC, LDS rows stride LDS_STRIDE) * W^T
// where W is row-major (out_ch, in_ch), i.e. B[k][n] = W[n][k].
// f32 16x16x4 layouts (ISA 7.12.2):
//   A: lane L -> M = L%16, VGPR0/1 hold K = kbase + (L/16)*2 + {0,1}
//   B: lane L -> N = L%16, VGPR0/1 hold K = kbase + (L/16)*2 + {0,1}
//   C/D: VGPR v -> lanes 0-15: (M=v, N=lane); lanes 16-31: (M=v+8, N=lane-16)
// ---------------------------------------------------------------------------
__device__ __forceinline__ v8f gemm_frag_wT(const float* __restrict__ As,
                                            const float* __restrict__ Wmat,
                                            int colbase, v8f acc) {
  const int lane  = threadIdx.x & 31;
  const int mrow  = lane & 15;
  const int khalf = (lane >> 4) << 1;
  const float* bp = Wmat + (size_t)(colbase + mrow) * CCH + khalf;
  #pragma unroll 4
  for (int k = 0; k < CCH; k += 4) {
    v2f a, b;
    a.x = As[mrow * LDS_STRIDE + k + khalf];
    a.y = As[mrow * LDS_STRIDE + k + khalf + 1];
    b.x = bp[k];
    b.y = bp[k + 1];
    acc = __builtin_amdgcn_wmma_f32_16x16x4_f32(false, a, false, b,
                                                (short)0, acc, false, false);
  }
  return acc;
}

// ---------------------------------------------------------------------------
// 1) Positional encoding add: xn[b][n][c] = x[b][c][h][w] + pe(c,h,w)
// ---------------------------------------------------------------------------
__global__ void posenc_kernel(const float* __restrict__ xin,
                              float* __restrict__ xn) {
  int i = blockIdx.x * blockDim.x + threadIdx.x;
  const int total = BATCH * NODES1 * CCH;
  if (i >= total) return;
  int c = i % CCH;
  int n = (i / CCH) % NODES1;
  int b = i / (CCH * NODES1);
  int h = n / WDIM, w = n % WDIM;
  float v = xin[((size_t)(b * CCH + c) * HDIM + h) * WDIM + w];
  int cc  = (c < 96) ? c : (c - 96);
  int pos = (c < 96) ? h : w;
  int jj  = cc >> 1;
  float invf = powf(10000.0f, -(float)(jj * 2) / 96.0f);
  float arg  = (float)pos * invf;
  float pe   = (cc & 1) ? cosf(arg) : sinf(arg);
  xn[i] = v + pe;
}

// ---------------------------------------------------------------------------
// 2) Row squared norms: SQ[b*N+n] = sum_c X[b][n][c]^2   (64 threads/node)
// ---------------------------------------------------------------------------
__global__ void rownorm_kernel(const float* __restrict__ X,
                               float* __restrict__ SQ, int N) {
  const int node = blockIdx.x;  // b*N + n
  const int t = threadIdx.x;
  const float* xp = X + (size_t)node * CCH;
  float s = 0.f;
  for (int c = t; c < CCH; c += 64) { float v = xp[c]; s += v * v; }
  __shared__ float red[64];
  red[t] = s;
  __syncthreads();
  for (int off = 32; off > 0; off >>= 1) {
    if (t < off) red[t] += red[t + off];
    __syncthreads();
  }
  if (t == 0) SQ[node] = red[0];
}

// ---------------------------------------------------------------------------
// 3) WMMA kNN: 16 query rows per block, 4 waves x 16-col subtiles, K=9 smallest
// ---------------------------------------------------------------------------
__global__ __launch_bounds__(128) void knn_kernel(const float* __restrict__ X,
                                                  const float* __restrict__ SQ,
                                                  int* __restrict__ IDX, int N) {
  const int b    = blockIdx.y;
  const int row0 = blockIdx.x * 16;
  const float* Xb  = X  + (size_t)b * N * CCH;
  const float* SQb = SQ + (size_t)b * N;
  const int t    = threadIdx.x;
  const int lane = t & 31;
  const int wave = t >> 5;

  __shared__ float As[16 * LDS_STRIDE];
  __shared__ float Ds[16 * 65];
  __shared__ float PV[16 * 8 * KNN_K];
  __shared__ int   PI[16 * 8 * KNN_K];

  for (int i = t; i < 16 * CCH; i += 128) {
    int r = i / CCH, c = i - r * CCH;
    As[r * LDS_STRIDE + c] = Xb[(size_t)(row0 + r) * CCH + c];
  }
  __syncthreads();

  const int mrow  = lane & 15;
  const int khalf = (lane >> 4) << 1;
  const int nn    = lane & 15;
  const int mofs  = (lane >> 4) << 3;

  float rsq[8];
  #pragma unroll
  for (int v = 0; v < 8; ++v) rsq[v] = SQb[row0 + mofs + v];

  float bv[KNN_K]; int bi[KNN_K];
  #pragma unroll
  for (int j = 0; j < KNN_K; ++j) { bv[j] = 3.0e38f; bi[j] = 0; }

  for (int cb = 0; cb < N; cb += 64) {
    const int colbase = cb + wave * 16;
    const float* Bp = Xb + (size_t)(colbase + mrow) * CCH + khalf;
    if (cb + 64 < N)  // hint next column block toward the caches
      __builtin_prefetch(Xb + (size_t)(cb + 64 + wave * 16 + mrow) * CCH, 0, 1);
    v8f acc = {};
    #pragma unroll 4
    for (int k = 0; k < CCH; k += 4) {
      v2f a, bf;
      a.x  = As[mrow * LDS_STRIDE + k + khalf];
      a.y  = As[mrow * LDS_STRIDE + k + khalf + 1];
      bf.x = Bp[k];
      bf.y = Bp[k + 1];
      acc = __builtin_amdgcn_wmma_f32_16x16x4_f32(false, a, false, bf,
                                                  (short)0, acc, false, false);
    }
    float csq = SQb[colbase + nn];
    #pragma unroll
    for (int v = 0; v < 8; ++v)
      Ds[(mofs + v) * 65 + wave * 16 + nn] = rsq[v] + csq - 2.0f * acc[v];
    __syncthreads();

    {  // top-9 insertion scan: thread t -> row t%16, column part t/16 (8 cols)
      const int r = t & 15, p = t >> 4;
      #pragma unroll
      for (int i = 0; i < 8; ++i) {
        int cc = p * 8 + i;
        float d = Ds[r * 65 + cc];
        if (d < bv[KNN_K - 1]) {
          int gi = cb + cc;
          int j = KNN_K - 1;
          while (j > 0 && bv[j - 1] > d) {
            bv[j] = bv[j - 1]; bi[j] = bi[j - 1]; --j;
          }
          bv[j] = d; bi[j] = gi;
        }
      }
    }
    __syncthreads();
  }

  {
    const int r = t & 15, p = t >> 4;
    #pragma unroll
    for (int j = 0; j < KNN_K; ++j) {
      PV[(r * 8 + p) * KNN_K + j] = bv[j];
      PI[(r * 8 + p) * KNN_K + j] = bi[j];
    }
  }
  __syncthreads();
  if (t < 16) {  // merge 8 partial sorted lists per row
    float mv[KNN_K]; int mi[KNN_K];
    #pragma unroll
    for (int j = 0; j < KNN_K; ++j) { mv[j] = 3.0e38f; mi[j] = 0; }
    for (int p = 0; p < 8; ++p)
      for (int j = 0; j < KNN_K; ++j) {
        float d = PV[(t * 8 + p) * KNN_K + j];
        if (d < mv[KNN_K - 1]) {
          int gi = PI[(t * 8 + p) * KNN_K + j];
          int jj = KNN_K - 1;
          while (jj > 0 && mv[jj - 1] > d) {
            mv[jj] = mv[jj - 1]; mi[jj] = mi[jj - 1]; --jj;
          }
          mv[jj] = d; mi[jj] = gi;
        }
      }
    int* op = IDX + ((size_t)b * N + row0 + t) * KNN_K;
    #pragma unroll
    for (int j = 0; j < KNN_K; ++j) op[j] = mi[j];
  }
}

// ---------------------------------------------------------------------------
// 4) Neighbor gather + mean: OUT[b][n][c] = mean_j X[b][idx[n][j]][c]
// ---------------------------------------------------------------------------
__global__ void gather_mean_kernel(const float* __restrict__ X,
                                   const int* __restrict__ IDX,
                                   float* __restrict__ OUT, int N) {
  const int node = blockIdx.x;     // b*N + n
  const int b = node / N;
  const int t = threadIdx.x;       // CCH threads
  __shared__ int nb[KNN_K];
  if (t < KNN_K) nb[t] = IDX[(size_t)node * KNN_K + t];
  __syncthreads();
  const float* Xb = X + (size_t)b * N * CCH;
  float s = 0.f;
  #pragma unroll
  for (int j = 0; j < KNN_K; ++j) s += Xb[(size_t)nb[j] * CCH + t];
  OUT[(size_t)node * CCH + t] = s * (1.0f / (float)KNN_K);
}

// ---------------------------------------------------------------------------
// 5) SAGE: OUT = NB @ Wl^T + bl + Xn @ Wr^T   (two accumulated WMMA GEMMs)
// ---------------------------------------------------------------------------
__global__ __launch_bounds__(128) void sage_kernel(
    const float* __restrict__ NB, const float* __restrict__ Xn,
    const float* __restrict__ Wl, const float* __restrict__ bl,
    const float* __restrict__ Wr, float* __restrict__ OUT, int N) {
  const int b = blockIdx.z;
  const int row0 = blockIdx.x * 16;
  const int col0 = blockIdx.y * 64;
  const int t = threadIdx.x, lane = t & 31, wave = t >> 5;
  __shared__ float Anb[16 * LDS_STRIDE];
  __shared__ float Ax[16 * LDS_STRIDE];
  const float* NBb = NB + ((size_t)b * N + row0) * CCH;
  const float* Xb  = Xn + ((size_t)b * N + row0) * CCH;
  for (int i = t; i < 16 * CCH; i += 128) {
    int r = i / CCH, c = i - r * CCH;
    Anb[r * LDS_STRIDE + c] = NBb[(size_t)r * CCH + c];
    Ax[r * LDS_STRIDE + c]  = Xb[(size_t)r * CCH + c];
  }
  __syncthreads();
  const int colbase = col0 + wave * 16;
  const int n = lane & 15, mofs = (lane >> 4) << 3;
  float bias = bl[colbase + n];
  v8f acc;
  #pragma unroll
  for (int v = 0; v < 8; ++v) acc[v] = bias;
  acc = gemm_frag_wT(Anb, Wl, colbase, acc);
  acc = gemm_frag_wT(Ax,  Wr, colbase, acc);
  float* Ob = OUT + ((size_t)b * N + row0) * CCH;
  #pragma unroll
  for (int v = 0; v < 8; ++v)
    Ob[(size_t)(mofs + v) * CCH + colbase + n] = acc[v];
}

// ---------------------------------------------------------------------------
// 6) Pool score: S[b*N+n] = tanh(dot(x_n, w) / ||w||)
// ---------------------------------------------------------------------------
__global__ void score_kernel(const float* __restrict__ X,
                             const float* __restrict__ w,
                             float* __restrict__ S, int N) {
  const int node = blockIdx.x;
  const int t = threadIdx.x;  // 64
  const float* xp = X + (size_t)node * CCH;
  float dv = 0.f, wn = 0.f;
  for (int c = t; c < CCH; c += 64) {
    float wv = w[c];
    dv += xp[c] * wv;
    wn += wv * wv;
  }
  __shared__ float r1[64], r2[64];
  r1[t] = dv; r2[t] = wn;
  __syncthreads();
  for (int off = 32; off > 0; off >>= 1) {
    if (t < off) { r1[t] += r1[t + off]; r2[t] += r2[t + off]; }
    __syncthreads();
  }
  if (t == 0) S[node] = tanhf(r1[0] * rsqrtf(r2[0]));
}

// ---------------------------------------------------------------------------
// 7) Per-batch bitonic sort (descending, index-stable) + keep top KEEP
// ---------------------------------------------------------------------------
__global__ __launch_bounds__(1024) void sort_topk_kernel(
    const float* __restrict__ S, float* __restrict__ TV,
    int* __restrict__ PM, int N, int KEEP) {
  const int b = blockIdx.x, t = threadIdx.x, nth = blockDim.x;
  __shared__ float V[4096];
  __shared__ int   I[4096];
  for (int i = t; i < N; i += nth) { V[i] = S[(size_t)b * N + i]; I[i] = i; }
  __syncthreads();
  for (int k = 2; k <= N; k <<= 1)
    for (int j = k >> 1; j > 0; j >>= 1) {
      for (int i = t; i < N; i += nth) {
        int ixj = i ^ j;
        if (ixj > i) {
          bool desc = ((i & k) == 0);
          float a = V[i], c = V[ixj];
          int ia = I[i], ic = I[ixj];
          bool greater = (a > c) || (a == c && ia < ic);
          bool doswap = desc ? (!greater) : greater;
          if (doswap) { V[i] = c; V[ixj] = a; I[i] = ic; I[ixj] = ia; }
        }
      }
      __syncthreads();
    }
  for (int i = t; i < KEEP; i += nth) {
    TV[(size_t)b * KEEP + i] = V[i];
    PM[(size_t)b * KEEP + i] = I[i];
  }
}

// ---------------------------------------------------------------------------
// 8) Gate + gather kept rows: OUT[b][i][c] = X[b][perm[i]][c] * topv[i]
// ---------------------------------------------------------------------------
__global__ void gate_gather_kernel(const float* __restrict__ X,
                                   const int* __restrict__ PM,
                                   const float* __restrict__ TV,
                                   float* __restrict__ OUT, int N, int KEEP) {
  const int i = blockIdx.x;        // b*KEEP + row
  const int b = i / KEEP;
  const int t = threadIdx.x;       // CCH
  int src = PM[i];
  float g = TV[i];
  OUT[(size_t)i * CCH + t] = X[((size_t)b * N + src) * CCH + t] * g;
}

// ---------------------------------------------------------------------------
// 9) GDN norm GEMM: NRM[n][o] = beta[o] + sum_c gamma[o][c] * x[n][c]^2
// ---------------------------------------------------------------------------
__global__ __launch_bounds__(128) void gdn_norm_kernel(
    const float* __restrict__ X, const float* __restrict__ gamma,
    const float* __restrict__ beta, float* __restrict__ NRM, int N) {
  const int b = blockIdx.z;
  const int row0 = blockIdx.x * 16;
  const int col0 = blockIdx.y * 64;
  const int t = threadIdx.x, lane = t & 31, wave = t >> 5;
  __shared__ float Asq[16 * LDS_STRIDE];
  const float* Xb = X + ((size_t)b * N + row0) * CCH;
  for (int i = t; i < 16 * CCH; i += 128) {
    int r = i / CCH, c = i - r * CCH;
    float v = Xb[(size_t)r * CCH + c];
    Asq[r * LDS_STRIDE + c] = v * v;
  }
  __syncthreads();
  const int colbase = col0 + wave * 16;
  const int n = lane & 15, mofs = (lane >> 4) << 3;
  float bias = beta[colbase + n];
  v8f acc;
  #pragma unroll
  for (int v = 0; v < 8; ++v) acc[v] = bias;
  acc = gemm_frag_wT(Asq, gamma, colbase, acc);
  float* Ob = NRM + ((size_t)b * N + row0) * CCH;
  #pragma unroll
  for (int v = 0; v < 8; ++v)
    Ob[(size_t)(mofs + v) * CCH + colbase + n] = acc[v];
}

__global__ void gdn_apply_kernel(const float* __restrict__ X,
                                 const float* __restrict__ NRM,
                                 float* __restrict__ Y, int total) {
  int i = blockIdx.x * blockDim.x + threadIdx.x;
  if (i >= total) return;
  Y[i] = X[i] * rsqrtf(NRM[i]);
}

// ---------------------------------------------------------------------------
// 10) Final NHWC(node-major) -> NCHW write
// ---------------------------------------------------------------------------
__global__ void final_kernel(const float* __restrict__ XP,
                             float* __restrict__ OUT) {
  int i = blockIdx.x * blockDim.x + threadIdx.x;
  const int total = BATCH * CCH * 16 * 16;
  if (i >= total) return;
  int w = i % 16, h = (i / 16) % 16, c = (i / 256) % CCH, b = i / (256 * CCH);
  OUT[i] = XP[((size_t)b * NODES3 + h * 16 + w) * CCH + c];
}

// ---------------------------------------------------------------------------
extern "C" void kernel_launch(void* const* d_in, const int* in_sizes, int n_in,
                              void* d_out, int out_size, void* d_ws, size_t ws_size,
                              hipStream_t stream) {
  const float* x      = (const float*)d_in[0];
  const float* W1l    = (const float*)d_in[1];
  const float* b1l    = (const float*)d_in[2];
  const float* W1r    = (const float*)d_in[3];
  const float* pool1w = (const float*)d_in[4];
  const float* gbeta  = (const float*)d_in[5];
  const float* ggamma = (const float*)d_in[6];
  const float* W2l    = (const float*)d_in[7];
  const float* b2l    = (const float*)d_in[8];
  const float* W2r    = (const float*)d_in[9];
  const float* pool2w = (const float*)d_in[10];
  float* out = (float*)d_out;

  // Workspace layout (~55 MB of f32)
  float* p = (float*)d_ws;
  float* xn1  = p; p += (size_t)BATCH * NODES1 * CCH;
  float* sq1  = p; p += (size_t)BATCH * NODES1;
  int*   idx1 = (int*)p; p += (size_t)BATCH * NODES1 * KNN_K;
  float* nbm1 = p; p += (size_t)BATCH * NODES1 * CCH;
  float* xs1  = p; p += (size_t)BATCH * NODES1 * CCH;
  float* sc1  = p; p += (size_t)BATCH * NODES1;
  float* tv1  = p; p += (size_t)BATCH * NODES2;
  int*   pm1  = (int*)p; p += (size_t)BATCH * NODES2;
  float* xp1  = p; p += (size_t)BATCH * NODES2 * CCH;
  float* nrm1 = p; p += (size_t)BATCH * NODES2 * CCH;
  float* xg   = p; p += (size_t)BATCH * NODES2 * CCH;
  float* sq2  = p; p += (size_t)BATCH * NODES2;
  int*   idx2 = (int*)p; p += (size_t)BATCH * NODES2 * KNN_K;
  float* nbm2 = p; p += (size_t)BATCH * NODES2 * CCH;
  float* xs2  = p; p += (size_t)BATCH * NODES2 * CCH;
  float* sc2  = p; p += (size_t)BATCH * NODES2;
  float* tv2  = p; p += (size_t)BATCH * NODES3;
  int*   pm2  = (int*)p; p += (size_t)BATCH * NODES3;
  float* xp2  = p; p += (size_t)BATCH * NODES3 * CCH;

  // Stage 1: pos-enc + norms + kNN + SAGE + pool
  {
    int total1 = BATCH * NODES1 * CCH;
    posenc_kernel<<<(total1 + 255) / 256, 256, 0, stream>>>(x, xn1);
    rownorm_kernel<<<BATCH * NODES1, 64, 0, stream>>>(xn1, sq1, NODES1);
    knn_kernel<<<dim3(NODES1 / 16, BATCH), 128, 0, stream>>>(xn1, sq1, idx1, NODES1);
    gather_mean_kernel<<<BATCH * NODES1, CCH, 0, stream>>>(xn1, idx1, nbm1, NODES1);
    sage_kernel<<<dim3(NODES1 / 16, CCH / 64, BATCH), 128, 0, stream>>>(
        nbm1, xn1, W1l, b1l, W1r, xs1, NODES1);
    score_kernel<<<BATCH * NODES1, 64, 0, stream>>>(xs1, pool1w, sc1, NODES1);
    sort_topk_kernel<<<BATCH, 1024, 0, stream>>>(sc1, tv1, pm1, NODES1, NODES2);
    gate_gather_kernel<<<BATCH * NODES2, CCH, 0, stream>>>(xs1, pm1, tv1, xp1,
                                                           NODES1, NODES2);
  }

  // Stage 2: GDN
  {
    gdn_norm_kernel<<<dim3(NODES2 / 16, CCH / 64, BATCH), 128, 0, stream>>>(
        xp1, ggamma, gbeta, nrm1, NODES2);
    int total2 = BATCH * NODES2 * CCH;
    gdn_apply_kernel<<<(total2 + 255) / 256, 256, 0, stream>>>(xp1, nrm1, xg, total2);
  }

  // Stage 3: recompute graph + SAGE + pool on 1024 nodes
  {
    rownorm_kernel<<<BATCH * NODES2, 64, 0, stream>>>(xg, sq2, NODES2);
    knn_kernel<<<dim3(NODES2 / 16, BATCH), 128, 0, stream>>>(xg, sq2, idx2, NODES2);
    gather_mean_kernel<<<BATCH * NODES2, CCH, 0, stream>>>(xg, idx2, nbm2, NODES2);
    sage_kernel<<<dim3(NODES2 / 16, CCH / 64, BATCH), 128, 0, stream>>>(
        nbm2, xg, W2l, b2l, W2r, xs2, NODES2);
    score_kernel<<<BATCH * NODES2, 64, 0, stream>>>(xs2, pool2w, sc2, NODES2);
    sort_topk_kernel<<<BATCH, 1024, 0, stream>>>(sc2, tv2, pm2, NODES2, NODES3);
    gate_gather_kernel<<<BATCH * NODES3, CCH, 0, stream>>>(xs2, pm2, tv2, xp2,
                                                           NODES2, NODES3);
  }

  // Final layout transform to (B, COUT, 16, 16)
  {
    int totf = BATCH * CCH * 16 * 16;
    final_kernel<<<(totf + 255) / 256, 256, 0, stream>>>(xp2, out);
  }
}